// SSDLoss_14525579395398
// MI455X (gfx1250) — compile-verified
//
#include <hip/hip_runtime.h>
#include <stdint.h>

// SSD anchor matching for MI455X (gfx1250).
// CDNA5-specific path: Tensor Data Mover (tensor_load_to_lds) stages the
// label/index tables into LDS once per workgroup; completion via TENSORcnt.

#define IGNORE_LBL (-100)
#define TAG_O_LBL  (1)

static constexpr int B_ = 16;
static constexpr int S_ = 2048;
static constexpr int A_ = 16;
static constexpr int E_ = 1024;
static constexpr int SLICES  = 16;          // split S across blocks
static constexpr int S_PER   = S_ / SLICES; // 128 rows of S per block
static constexpr int THREADS = 256;         // 8 wave32s

typedef uint32_t v4u32 __attribute__((ext_vector_type(4)));
typedef int      v8i32 __attribute__((ext_vector_type(8)));
typedef int      v4i32 __attribute__((ext_vector_type(4)));

// Build a CDNA5 Tensor DMA descriptor for a flat 1-D run of `n` dwords and
// issue TENSOR_LOAD_TO_LDS. D# layout per CDNA5 ISA ch.8:
//  group0: [1:0]=count=1, [63:32]=lds_addr, [120:64]=global_addr, [127:126]=type(2)
//  group1: [17:16]=data_size(2 -> 4B), tensor_dim0[79:48]=n, tensor_dim1[111:80]=1,
//          tile_dim0[127:112]=n, tile_dim1[143:128]=1, tensor_dim0_stride[207:160]=n
__device__ __forceinline__ void tdm_load_dwords_to_lds(uint32_t lds_off,
                                                       const void* gsrc,
                                                       uint32_t n) {
  uint64_t ga = (uint64_t)(uintptr_t)gsrc;
  v4u32 g0;
  g0[0] = 1u;                                                    // count = 1 (valid, user mode)
  g0[1] = lds_off;                                               // lds_addr (bytes)
  g0[2] = (uint32_t)ga;                                          // global_addr[31:0]
  g0[3] = (uint32_t)((ga >> 32) & 0x01FFFFFFull) | 0x80000000u;  // global_addr[56:32] | type=2
  v8i32 g1;
  g1[0] = 0x00020000;                                  // data_size = 2 (4 bytes), no mask/pad/iter
  g1[1] = (int)((n & 0xFFFFu) << 16);                  // tensor_dim0[15:0] @ bits 63:48
  g1[2] = (int)(((n >> 16) & 0xFFFFu) | (1u << 16));   // tensor_dim0[31:16] | tensor_dim1 = 1
  g1[3] = (int)((n & 0xFFFFu) << 16);                  // tensor_dim1 hi = 0 | tile_dim0 = n
  g1[4] = 1;                                           // tile_dim1 = 1, tile_dim2 = 0
  g1[5] = (int)n;                                      // tensor_dim0_stride[31:0] = n
  g1[6] = 0;                                           // stride hi / dim1_stride lo
  g1[7] = 0;                                           // dim1_stride hi
  v4i32 z4 = {0, 0, 0, 0};
#if defined(__clang_major__) && (__clang_major__ >= 23)
  v8i32 z8 = {0, 0, 0, 0, 0, 0, 0, 0};
  __builtin_amdgcn_tensor_load_to_lds(g0, g1, z4, z4, z8, 0);
#else
  __builtin_amdgcn_tensor_load_to_lds(g0, g1, z4, z4, 0);
#endif
}

__global__ __launch_bounds__(THREADS)
void ssd_match_kernel(const int* __restrict__ input_len,   // (B)
                      const int* __restrict__ label,       // (E,3) int32: type,left,right
                      const int* __restrict__ index,       // (E) sorted int32 batch ids
                      const float* __restrict__ pos_t_p,   // scalar
                      const float* __restrict__ neg_t_p,   // scalar
                      int* __restrict__ out_cls,           // (B,S,A) int32
                      float* __restrict__ out_iou) {       // (B,S,A) f32
  __shared__ int   lds_label[E_ * 3];  // raw (type,left,right) from TDM
  __shared__ int   lds_index[E_];      // raw batch ids from TDM
  __shared__ float lds_L[E_];          // left + 1
  __shared__ float lds_W[E_];          // width (right - left)
  __shared__ int   lds_T[E_];          // type

  const int b   = blockIdx.x;
  const int s0  = blockIdx.y * S_PER;
  const int tid = threadIdx.x;

  __builtin_prefetch(input_len, 0, 0);   // global_prefetch_b8

  // Wave 0 issues both TDM descriptors (TDM ignores EXEC -> one op per wave),
  // drains TENSORcnt, then the workgroup barrier publishes LDS to all waves.
  if (tid < 32) {
    tdm_load_dwords_to_lds((uint32_t)(uintptr_t)&lds_label[0], label, (uint32_t)(E_ * 3));
    tdm_load_dwords_to_lds((uint32_t)(uintptr_t)&lds_index[0], index, (uint32_t)E_);
    __builtin_amdgcn_s_wait_tensorcnt(0);
  }
  __syncthreads();

  // Unpack boxes: boxes = label[:,1:] + 1.0 ; all values exact in fp32.
  for (int e = tid; e < E_; e += THREADS) {
    const int t = lds_label[3 * e + 0];
    const int l = lds_label[3 * e + 1];
    const int r = lds_label[3 * e + 2];
    lds_T[e] = t;
    lds_L[e] = (float)l + 1.0f;
    lds_W[e] = (float)(r - l);
  }
  __syncthreads();

  // index is sorted: binary-search this batch's contiguous event range [e0,e1).
  int lo = 0, hi = E_;
  while (lo < hi) { const int m = (lo + hi) >> 1; if (lds_index[m] < b) lo = m + 1; else hi = m; }
  const int e0 = lo;
  hi = E_;
  while (lo < hi) { const int m = (lo + hi) >> 1; if (lds_index[m] < b + 1) lo = m + 1; else hi = m; }
  const int e1 = lo;

  const int   ilen = input_len[b];
  const float post = *pos_t_p;
  const float negt = *neg_t_p;

  for (int f = tid; f < S_PER * A_; f += THREADS) {
    const int s = s0 + (f >> 4);     // A_ == 16
    const int a = f & (A_ - 1);
    const float sz = (float)(a + 1);
    const float la = ((float)s + 0.5f) - 0.5f * sz;
    const float ra = la + sz;

    // Running max with strict '>' keeps the earliest event among ties,
    // which equals segment_min over {e : iou == seg_max}. Empty range:
    // best = -inf (segment_max identity), barg clipped to E-1.
    float best = -__builtin_inff();
    int   barg = E_ - 1;
    for (int e = e0; e < e1; ++e) {
      const float lb = lds_L[e];        // broadcast LDS reads (all lanes same addr)
      const float w  = lds_W[e];
      const float rb = lb + w;
      float inter = fminf(ra, rb) - fmaxf(la, lb);
      inter = fmaxf(inter, 0.0f);
      const float iou = inter / ((sz + w) - inter);  // u > 0 always
      if (iou > best) { best = iou; barg = e; }
    }

    const bool pos = (s < ilen);
    int cls = IGNORE_LBL;
    if (best < negt) cls = TAG_O_LBL;
    if (best > post) cls = lds_T[barg];
    if (!pos)        cls = IGNORE_LBL;

    const size_t o = (size_t)b * (S_ * A_) + (size_t)s * A_ + (size_t)a;
    out_cls[o] = cls;
    out_iou[o] = pos ? best : 0.0f;
  }
}

extern "C" void kernel_launch(void* const* d_in, const int* in_sizes, int n_in,
                              void* d_out, int out_size, void* d_ws, size_t ws_size,
                              hipStream_t stream) {
  (void)in_sizes; (void)n_in; (void)out_size; (void)d_ws; (void)ws_size;
  // setup_inputs() order:
  // 0: input_len (B,i32)  1: cls_output (unused)  2: reg_output (unused)
  // 3: label (E,3,i32)    4: index (E,i32)        5: anchors (unused, rebuilt exactly)
  // 6: iou_thresh_pos f32 7: iou_thresh_neg f32
  const int*   input_len = (const int*)d_in[0];
  const int*   label     = (const int*)d_in[3];
  const int*   index     = (const int*)d_in[4];
  const float* pos_t     = (const float*)d_in[6];
  const float* neg_t     = (const float*)d_in[7];

  int*   out_cls = (int*)d_out;                            // cls_label, raw int32
  float* out_iou = (float*)d_out + (size_t)B_ * S_ * A_;   // ious_max, f32

  dim3 grid(B_, SLICES);
  ssd_match_kernel<<<grid, THREADS, 0, stream>>>(input_len, label, index,
                                                 pos_t, neg_t, out_cls, out_iou);
}